// LlamaCompressedMLP_30425548325397
// MI455X (gfx1250) — compile-verified
//
#include <hip/hip_runtime.h>
#include <stdint.h>

// LLaMA MLP: out = (silu(x@Wg^T) * (x@Wu^T)) @ Wd^T
// Shapes: x[4096,4096], w_gate_up[22016,4096], w_down[4096,11008], out[4096,4096]
// Memory-bound (~1.03 GB f32 traffic vs 1.11 TFLOP => ~44us HBM floor at
// 23.3 TB/s, needing only ~25 TFLOPS). Keep f32 end-to-end and use
// V_WMMA_F32_16X16X4_F32; hide HBM latency with CDNA5 async global->LDS
// copies (ASYNCcnt) and double-buffered LDS tiles.

typedef __attribute__((ext_vector_type(2))) float v2f;
typedef __attribute__((ext_vector_type(8))) float v8f;

#define THREADS 256
#define BM 128
#define BN 64
#define BK 16
#define BKP 20   // padded K stride: 16B-aligned async LDS writes, conflict-free b64 reads

// 16-byte async copy global -> LDS, tracked by ASYNCcnt, no VGPR staging.
__device__ __forceinline__ void async_copy16(float* lds_dst, const float* gsrc) {
    uint32_t loff = (uint32_t)(uintptr_t)lds_dst;   // LDS aperture: addr[31:0]
    asm volatile("global_load_async_to_lds_b128 %0, %1, off"
                 :: "v"(loff), "v"(gsrc) : "memory");
}

__device__ __forceinline__ void wait_async0() {
#if __has_builtin(__builtin_amdgcn_s_wait_asynccnt)
    __builtin_amdgcn_s_wait_asynccnt(0);
#else
    asm volatile("s_wait_asynccnt 0x0" ::: "memory");
#endif
}

// C[M,N] = A[M,K] * B[N,K]^T ; if SWIGLU, two B-operands (gate rows Bg,
// up rows Bu), writes silu(gate)*up.
template<bool SWIGLU>
__global__ __launch_bounds__(THREADS)
void gemm_bt_wmma(const float* __restrict__ A,
                  const float* __restrict__ Bg,
                  const float* __restrict__ Bu,
                  float* __restrict__ C,
                  int N, int K)
{
    __shared__ float sA[2][BM][BKP];
    __shared__ float sB[2][SWIGLU ? 2 : 1][BN][BKP];

    const int m0 = blockIdx.x * BM;
    const int n0 = blockIdx.y * BN;

    const int tid  = threadIdx.x;
    const int lane = tid & 31;
    const int ll   = lane & 15;   // M/N index within 16-wide tile
    const int lh   = lane >> 4;   // selects K-pair (A/B) / row-half (C)
    const int wid  = tid >> 5;    // 8 waves
    const int wm   = (wid & 3) * 32;   // wave row offset: 4 x 32 = 128
    const int wn   = (wid >> 2) * 32;  // wave col offset: 2 x 32 = 64

    // Per-thread async fill of one LDS buffer (BK=16 -> 2 x b128 for A, 1 each for B).
    auto fill = [&](int buf, int k0) {
#pragma unroll
        for (int i = 0; i < (BM * BK) / (THREADS * 4); ++i) {   // 2
            int idx = tid + i * THREADS;
            int r = idx >> 2;            // BK/4 = 4 float4 per row
            int c = (idx & 3) * 4;
            async_copy16(&sA[buf][r][c], A + (size_t)(m0 + r) * K + k0 + c);
        }
#pragma unroll
        for (int i = 0; i < (BN * BK) / (THREADS * 4); ++i) {   // 1
            int idx = tid + i * THREADS;
            int r = idx >> 2;
            int c = (idx & 3) * 4;
            async_copy16(&sB[buf][0][r][c], Bg + (size_t)(n0 + r) * K + k0 + c);
            if constexpr (SWIGLU)
                async_copy16(&sB[buf][1][r][c], Bu + (size_t)(n0 + r) * K + k0 + c);
        }
    };

    const v8f vzero = {0.f,0.f,0.f,0.f,0.f,0.f,0.f,0.f};
    v8f accG[2][2];
    v8f accU[2][2];
#pragma unroll
    for (int rt = 0; rt < 2; ++rt)
#pragma unroll
        for (int ct = 0; ct < 2; ++ct) { accG[rt][ct] = vzero; accU[rt][ct] = vzero; }

    const int nK = K / BK;
    fill(0, 0);                              // prologue fetch

    for (int kb = 0; kb < nK; ++kb) {
        const int buf = kb & 1;
        wait_async0();        // my async writes into sX[buf] have landed in LDS
        __syncthreads();      // everyone's landed; everyone done reading buf^1
        if (kb + 1 < nK)
            fill(buf ^ 1, (kb + 1) * BK);    // overlaps with compute below

        // ---- WMMA inner loop: K in steps of 4 ----
#pragma unroll
        for (int kk = 0; kk < BK; kk += 4) {
            const int col = kk + 2 * lh;     // per-lane K pair (even -> 8B aligned)
            v2f a[2], bg[2], bu[2];
#pragma unroll
            for (int rt = 0; rt < 2; ++rt)
                a[rt] = *(const v2f*)&sA[buf][wm + rt * 16 + ll][col];
#pragma unroll
            for (int ct = 0; ct < 2; ++ct) {
                bg[ct] = *(const v2f*)&sB[buf][0][wn + ct * 16 + ll][col];
                if constexpr (SWIGLU)
                    bu[ct] = *(const v2f*)&sB[buf][1][wn + ct * 16 + ll][col];
            }
#pragma unroll
            for (int rt = 0; rt < 2; ++rt)
#pragma unroll
                for (int ct = 0; ct < 2; ++ct) {
                    accG[rt][ct] = __builtin_amdgcn_wmma_f32_16x16x4_f32(
                        false, a[rt], false, bg[ct], (short)0, accG[rt][ct],
                        false, false);
                    if constexpr (SWIGLU)
                        accU[rt][ct] = __builtin_amdgcn_wmma_f32_16x16x4_f32(
                            false, a[rt], false, bu[ct], (short)0, accU[rt][ct],
                            false, false);
                }
        }
    }

    // ---- epilogue: C/D layout row = vgpr + 8*(lane/16), col = lane%16 ----
#pragma unroll
    for (int rt = 0; rt < 2; ++rt)
#pragma unroll
        for (int ct = 0; ct < 2; ++ct) {
            const int col = n0 + wn + ct * 16 + ll;
#pragma unroll
            for (int v = 0; v < 8; ++v) {
                const int row = m0 + wm + rt * 16 + v + 8 * lh;
                float g = accG[rt][ct][v];
                float r;
                if constexpr (SWIGLU) {
                    float u = accU[rt][ct][v];
                    r = (g / (1.0f + __expf(-g))) * u;   // silu(g) * u
                } else {
                    r = g;
                }
                C[(size_t)row * N + col] = r;
            }
        }
}

extern "C" void kernel_launch(void* const* d_in, const int* in_sizes, int n_in,
                              void* d_out, int out_size, void* d_ws, size_t ws_size,
                              hipStream_t stream) {
    (void)in_sizes; (void)n_in; (void)out_size; (void)ws_size;

    const float* x   = (const float*)d_in[0];   // [4096, 4096]
    const float* wgu = (const float*)d_in[1];   // [22016, 4096]
    const float* wd  = (const float*)d_in[2];   // [4096, 11008]
    float*       out = (float*)d_out;           // [4096, 4096]
    float*       h   = (float*)d_ws;            // [4096, 11008] = 180.4 MB scratch

    const int H = 4096, I = 11008;
    const int M = 2 * 2048;                     // B*S = 4096 tokens

    // GEMM1 + SwiGLU: h = silu(x@Wg^T) * (x@Wu^T).
    // grid.x = M tiles (fastest) so concurrent blocks share the same weight
    // strip through the 192MB L2; x (64MB) stays L2-resident.
    dim3 g1(M / BM, I / BN);                    // (32, 172)
    gemm_bt_wmma<true><<<g1, THREADS, 0, stream>>>(
        x, wgu, wgu + (size_t)I * H, h, I, H);

    // GEMM2: out = h @ w_down^T
    dim3 g2(M / BM, H / BN);                    // (32, 64)
    gemm_bt_wmma<false><<<g2, THREADS, 0, stream>>>(
        h, wd, nullptr, out, H, I);
}